// CorpBEVT_42803644072541
// MI455X (gfx1250) — compile-verified
//
#include <hip/hip_runtime.h>
#include <stdint.h>

// Problem constants (fixed by the reference)
#define C_DIM   128
#define HW      65536          // 256*256
#define N_CAV   5
#define K_SEL   32768
#define MASK_WORDS (HW / 32)   // 2048 u32 = 8 KB
#define CHW4    (C_DIM * (HW / 4))   // float4 elements per cav = 2,097,152

typedef float v4f __attribute__((ext_vector_type(4)));

// ---------------------------------------------------------------------------
// Kernel 1: zero the selection bitmask (d_ws is poisoned with 0xAA by harness)
// ---------------------------------------------------------------------------
__global__ void corpbevt_zero_mask(unsigned int* __restrict__ mask) {
    int i = blockIdx.x * blockDim.x + threadIdx.x;
    if (i < MASK_WORDS) mask[i] = 0u;
}

// ---------------------------------------------------------------------------
// Kernel 2: set one bit per selected index (indices are unique)
// ---------------------------------------------------------------------------
__global__ void corpbevt_set_mask(const int* __restrict__ idx,
                                  unsigned int* __restrict__ mask) {
    int i = blockIdx.x * blockDim.x + threadIdx.x;
    if (i < K_SEL) {
        int p = idx[i];
        atomicOr(&mask[p >> 5], 1u << (p & 31));
    }
}

// ---------------------------------------------------------------------------
// Kernel 3: masked select, fully coalesced v4f streaming.
// out[n,c,p] = bit(p) ? x[n,c,p] : ego[c,p]
// - 8 KB bitmask staged into LDS via CDNA5 async global->LDS (ASYNCcnt)
// - x / ego / out streams use non-temporal cache policy (single-use data)
// ---------------------------------------------------------------------------
__global__ __launch_bounds__(256) void corpbevt_select(
    const v4f* __restrict__ x4,           // [N_CAV, C, HW/4]  (B=1, first N of L)
    const v4f* __restrict__ bev4,         // [N_CAV, C, HW/4]
    const int* __restrict__ ego_index,    // scalar on device
    const unsigned int* __restrict__ gmask, // [MASK_WORDS]
    v4f* __restrict__ out4)               // [N_CAV, C, HW/4]
{
    __shared__ __align__(16) unsigned int s_mask[MASK_WORDS];

    // LDS byte offset of s_mask (addrspace(3) pointers are 32-bit LDS offsets)
    uint32_t lds_base = (uint32_t)(uintptr_t)
        (__attribute__((address_space(3))) unsigned int*)&s_mask[0];

    const char* gbase = (const char*)gmask;
    int t = threadIdx.x;

    // 8 KB = 256 threads * 2 rounds * 16 B, via async copy (tracked by ASYNCcnt)
#pragma unroll
    for (int r = 0; r < 2; ++r) {
        uint32_t off      = (uint32_t)(r * 4096 + t * 16);
        uint32_t lds_addr = lds_base + off;
        const char* gaddr = gbase + off;
        asm volatile("global_load_async_to_lds_b128 %0, %1, off"
                     :: "v"(lds_addr), "v"(gaddr)
                     : "memory");
    }
#if __has_builtin(__builtin_amdgcn_s_wait_asynccnt)
    __builtin_amdgcn_s_wait_asynccnt(0);
#else
    asm volatile("s_wait_asynccnt 0" ::: "memory");
#endif
    __syncthreads();

    int g = blockIdx.x * 256 + t;          // v4f index into [C, HW/4]
    if (g >= CHW4) return;

    int p = (g & (HW / 4 - 1)) * 4;        // spatial base of this v4f
    unsigned int bits = s_mask[p >> 5] >> (p & 31);

    int e = ego_index[0];
    v4f ego = __builtin_nontemporal_load(&bev4[e * CHW4 + g]); // single-use stream

#pragma unroll
    for (int n = 0; n < N_CAV; ++n) {
        v4f xv = __builtin_nontemporal_load(&x4[n * CHW4 + g]);
        v4f o;
        o.x = (bits & 1u) ? xv.x : ego.x;
        o.y = (bits & 2u) ? xv.y : ego.y;
        o.z = (bits & 4u) ? xv.z : ego.z;
        o.w = (bits & 8u) ? xv.w : ego.w;
        __builtin_nontemporal_store(o, &out4[n * CHW4 + g]);
    }
}

// ---------------------------------------------------------------------------
// Host-side launcher
// ---------------------------------------------------------------------------
extern "C" void kernel_launch(void* const* d_in, const int* in_sizes, int n_in,
                              void* d_out, int out_size, void* d_ws, size_t ws_size,
                              hipStream_t stream) {
    const v4f* x4      = (const v4f*)d_in[0];     // [1,5,128,65536] f32
    const v4f* bev4    = (const v4f*)d_in[1];     // [5,128,256,256] f32
    const int* sel     = (const int*)d_in[2];     // [32768] i32
    const int* ego     = (const int*)d_in[3];     // [1] i32
    v4f* out4          = (v4f*)d_out;             // [5,128,256,256] f32
    unsigned int* mask = (unsigned int*)d_ws;     // 8 KB bitmask

    (void)in_sizes; (void)n_in; (void)out_size; (void)ws_size;

    corpbevt_zero_mask<<<(MASK_WORDS + 255) / 256, 256, 0, stream>>>(mask);
    corpbevt_set_mask<<<(K_SEL + 255) / 256, 256, 0, stream>>>(sel, mask);
    corpbevt_select<<<CHW4 / 256, 256, 0, stream>>>(x4, bev4, ego, mask, out4);
}